// _ScaledDotProductAttention_37666863186485
// MI455X (gfx1250) — compile-verified
//
#include <hip/hip_runtime.h>
#include <cstdint>

// Fused scaled-dot-product attention for MI455X (gfx1250, wave32, WMMA).
// reference: scores = (Q·K) * mask * scale + prev ; W = softmax(scores) ; O = W·V
// Outputs concatenated: [O | W | scores].

typedef __attribute__((ext_vector_type(2))) float v2f;
typedef __attribute__((ext_vector_type(8))) float v8f;

#define B_ 8
#define H_ 16
#define S_ 1024
#define D_ 64

#define SC_STRIDE 1025   // 1024 + 1 pad: break LDS bank conflicts
#define QS_STRIDE 66     // 64 + 2 pad

__global__ __launch_bounds__(128)
void fused_attn_kernel(const float* __restrict__ q,
                       const float* __restrict__ k,
                       const float* __restrict__ v,
                       const float* __restrict__ prev,
                       const float* __restrict__ mask,
                       const float* __restrict__ scale_p,
                       float* __restrict__ outO,
                       float* __restrict__ outW,
                       float* __restrict__ outS)
{
    extern __shared__ float smem[];
    float* sc   = smem;                      // [16][SC_STRIDE] scores -> exp -> weights
    float* qs   = sc + 16 * SC_STRIDE;       // [16][QS_STRIDE] Q tile
    float* red  = qs + 16 * QS_STRIDE;       // [16][8] partial reductions
    float* rowv = red + 16 * 8;              // [16] row max, then row inv-sum

    const int tid  = threadIdx.x;
    const int lane = tid & 31;
    const int wave = tid >> 5;               // 0..3

    const int bid = blockIdx.x;
    const int qt  = bid & 63;                // q-tile within (b,h): S/16 = 64
    const int bh  = bid >> 6;                // 0..127
    const int q0  = qt * 16;

    const float scale = scale_p[0];

    const size_t kBase = (size_t)bh * D_ * S_;            // k[bh][d][s]  (D x S)
    const size_t qBase = ((size_t)bh * S_ + q0) * D_;     // q[bh][q0..][d]
    const size_t vBase = (size_t)bh * S_ * D_;            // v[bh][s][d]
    const size_t pBase = ((size_t)bh * S_ + q0) * S_;     // prev row-block
    const size_t mBase = (size_t)q0 * S_;                 // mask row-block (broadcast b,h)

    // ---- load Q tile (16 x 64) to LDS, coalesced ----
    for (int i = tid; i < 16 * D_; i += 128) {
        const int r = i >> 6, c = i & 63;
        qs[r * QS_STRIDE + c] = q[qBase + (size_t)r * D_ + c];
    }
    __syncthreads();

    // lane mapping for 16x16x4 f32 WMMA fragments (wave32):
    //   A: lanes 0-15 hold M=lane, K={k0,k0+1}; lanes 16-31 hold M=lane-16, K={k0+2,k0+3}
    //   C/D: VGPR e holds M = e + (lane<16 ? 0 : 8), N = lane%16
    const int halfSel = (lane < 16) ? 0 : 2;
    const int ln      = lane & 15;

    // ---- preload all A fragments of the Q tile (16 x 64) into registers ----
    v2f afrag[16];
#pragma unroll
    for (int kk = 0; kk < 16; ++kk) {
        const int col = kk * 4 + halfSel;
        afrag[kk].x = qs[ln * QS_STRIDE + col];
        afrag[kk].y = qs[ln * QS_STRIDE + col + 1];
    }

    // ---- phase 1: scores = (Q·K) * mask * scale + prev ; each wave owns 256 cols ----
    const int n0base = wave * 256;
    for (int nt = 0; nt < 16; ++nt) {
        const int n0  = n0base + nt * 16;
        const int col = n0 + ln;
        v8f c = {0.f, 0.f, 0.f, 0.f, 0.f, 0.f, 0.f, 0.f};
#pragma unroll
        for (int kk = 0; kk < 16; ++kk) {
            const int d0 = kk * 4 + halfSel;
            v2f b;
            b.x = k[kBase + (size_t)d0 * S_ + col];
            b.y = k[kBase + (size_t)(d0 + 1) * S_ + col];
            c = __builtin_amdgcn_wmma_f32_16x16x4_f32(
                    false, afrag[kk], false, b, (short)0, c, false, false);
        }
#pragma unroll
        for (int e = 0; e < 8; ++e) {
            const int m = e + (halfSel << 2);  // +8 for lanes 16-31
            const float val = c[e] * mask[mBase + (size_t)m * S_ + col] * scale
                            + prev[pBase + (size_t)m * S_ + col];
            sc[m * SC_STRIDE + col] = val;
            outS[pBase + (size_t)m * S_ + col] = val;   // stream scores once
        }
    }

    // prefetch V rows toward caches while softmax runs (gfx1250 global_prefetch)
    for (int s = tid; s < S_; s += 128) {
        __builtin_prefetch(&v[vBase + (size_t)s * D_], 0, 1);
        __builtin_prefetch(&v[vBase + (size_t)s * D_ + 32], 0, 1);
    }
    __syncthreads();

    // ---- phase 2: row softmax over 1024 cols; 8 threads per row ----
    {
        const int r     = tid & 15;
        const int ch    = tid >> 4;          // 0..7
        const int cbase = ch * 128;

        float lmax = -3.402823466e38f;
        for (int i = 0; i < 128; ++i)
            lmax = fmaxf(lmax, sc[r * SC_STRIDE + cbase + i]);
        red[r * 8 + ch] = lmax;
        __syncthreads();
        if (ch == 0) {
            float m = red[r * 8];
#pragma unroll
            for (int j = 1; j < 8; ++j) m = fmaxf(m, red[r * 8 + j]);
            rowv[r] = m;
        }
        __syncthreads();

        const float rmax = rowv[r];
        float lsum = 0.f;
        for (int i = 0; i < 128; ++i) {
            const float e = __expf(sc[r * SC_STRIDE + cbase + i] - rmax);
            sc[r * SC_STRIDE + cbase + i] = e;
            lsum += e;
        }
        red[r * 8 + ch] = lsum;
        __syncthreads();
        if (ch == 0) {
            float s = 0.f;
#pragma unroll
            for (int j = 0; j < 8; ++j) s += red[r * 8 + j];
            rowv[r] = 1.0f / s;
        }
        __syncthreads();
    }

    // normalize in LDS + coalesced write of weights (consecutive lanes -> consecutive cols)
    for (int i = tid; i < 16 * 1024; i += 128) {
        const int r = i >> 10, cidx = i & 1023;
        const float w = sc[r * SC_STRIDE + cidx] * rowv[r];
        sc[r * SC_STRIDE + cidx] = w;
        outW[pBase + (size_t)r * S_ + cidx] = w;
    }
    __syncthreads();

    // ---- phase 3: O = W (16x1024, LDS) x V (1024x64, global); wave owns 16 d-cols ----
    {
        const int dcol = wave * 16 + ln;
        v8f c = {0.f, 0.f, 0.f, 0.f, 0.f, 0.f, 0.f, 0.f};
#pragma unroll 4
        for (int kt = 0; kt < 256; ++kt) {
            const int s0 = kt * 4 + halfSel;
            v2f a, b;
            a.x = sc[ln * SC_STRIDE + s0];
            a.y = sc[ln * SC_STRIDE + s0 + 1];
            b.x = v[vBase + (size_t)s0 * D_ + dcol];
            b.y = v[vBase + (size_t)(s0 + 1) * D_ + dcol];
            c = __builtin_amdgcn_wmma_f32_16x16x4_f32(
                    false, a, false, b, (short)0, c, false, false);
        }
        const size_t oBase = ((size_t)bh * S_ + q0) * D_;
#pragma unroll
        for (int e = 0; e < 8; ++e) {
            const int m = e + (halfSel << 2);
            outO[oBase + (size_t)m * D_ + dcol] = c[e];
        }
    }
}

extern "C" void kernel_launch(void* const* d_in, const int* in_sizes, int n_in,
                              void* d_out, int out_size, void* d_ws, size_t ws_size,
                              hipStream_t stream) {
    (void)in_sizes; (void)n_in; (void)out_size; (void)d_ws; (void)ws_size;

    const float* q     = (const float*)d_in[0];
    const float* k     = (const float*)d_in[1];
    const float* v     = (const float*)d_in[2];
    const float* prev  = (const float*)d_in[3];
    const float* mask  = (const float*)d_in[4];
    const float* scale = (const float*)d_in[5];

    float* out = (float*)d_out;
    const size_t nO = (size_t)B_ * H_ * S_ * D_;   // output elems
    const size_t nW = (size_t)B_ * H_ * S_ * S_;   // weights elems
    float* outO = out;
    float* outW = out + nO;
    float* outS = outW + nW;

    const dim3 grid(B_ * H_ * (S_ / 16));          // 8192 workgroups
    const dim3 block(128);                         // 4 waves (wave32)
    const size_t smemBytes =
        (size_t)(16 * SC_STRIDE + 16 * QS_STRIDE + 16 * 8 + 16) * sizeof(float);

    fused_attn_kernel<<<grid, block, smemBytes, stream>>>(
        q, k, v, prev, mask, scale, outO, outW, outS);
}